// MultiHeadSpatialAttention_31782757990769
// MI455X (gfx1250) — compile-verified
//
#include <hip/hip_runtime.h>
#include <hip/hip_bf16.h>
#include <math.h>

// ---------------------------------------------------------------------------
// MultiHeadSpatialAttention for MI455X (gfx1250, wave32, WMMA)
// Shapes: B=2, C=256, HW=4096, heads=4, rhd=8, head_dim=64, qkv rows=320
// Full fp32 fidelity via V_WMMA_F32_16X16X4_F32.
// All indexing 32-bit (max per-batch buffer = 320*4096 floats << 2^31).
// ---------------------------------------------------------------------------

typedef float v2f __attribute__((ext_vector_type(2)));
typedef float v8f __attribute__((ext_vector_type(8)));

#define HW    4096
#define NH    4
#define RHD   8
#define QDIM  32          // rhd * nh
#define HD    64          // head_dim
#define CIN   256
#define OQKV  320         // (rhd*2 + head_dim) * nh
#define RSQRT8 0.35355339059327373f

__device__ __forceinline__ v8f wmma_f32_16x16x4(v2f a, v2f b, v8f c) {
  // D = A(16x4) * B(4x16) + C(16x16), fp32 matrix pipe
  return __builtin_amdgcn_wmma_f32_16x16x4_f32(false, a, false, b, (short)0, c,
                                               false, false);
}

// ---------------------------------------------------------------------------
// Kernel 1/4: Y[b,o,n] = W[o,:CIN] . X[b,:,n] + bias[o]; FUSED epilogue does
// Y = gamma*(acc+bias) + residual for the final projection.
// Block = 256 threads = 8 waves; block tile = 32(o) x 64(n); each wave owns
// one 16x16 output tile. K-loop over CIN in steps of 4 (one WMMA per step),
// unrolled 8x -> 32-element K slabs with clause-grouped loads.
// ---------------------------------------------------------------------------
template <bool FUSED>
__global__ __launch_bounds__(256) void gemm_bias_kernel(
    const float* __restrict__ W, const float* __restrict__ X,
    const float* __restrict__ bias, float* __restrict__ Y,
    int xbs, int ybs,
    const float* __restrict__ residual, const float* __restrict__ gamma)
{
  const int tid  = threadIdx.x;
  const int wave = tid >> 5;
  const int lane = tid & 31;
  const int hi   = lane >> 4;    // lane-half (selects K pair / M+8)
  const int l16  = lane & 15;
  const int b    = blockIdx.z;
  const int o0   = blockIdx.y * 32 + (wave >> 2) * 16;
  const int n0   = blockIdx.x * 64 + (wave & 3) * 16;

  // A rows: M = l16 (both halves); per-lane K pair = {2*hi, 2*hi+1}
  const float* wrow = W + (o0 + l16) * CIN + 2 * hi;
  // B rows: K rows {2*hi, 2*hi+1}, N = l16
  const float* xcol = X + b * xbs + (2 * hi) * HW + n0 + l16;

  v8f acc = {};
#pragma unroll 8
  for (int k = 0; k < CIN; k += 4) {
    v2f a = *(const v2f*)(wrow + k);
    v2f bb;
    bb.x = xcol[k * HW];
    bb.y = xcol[k * HW + HW];
    acc = wmma_f32_16x16x4(a, bb, acc);
  }

  float* Yb = Y + b * ybs;
  const float g = FUSED ? gamma[0] : 1.0f;
  const float* resb = FUSED ? (residual + b * ybs) : nullptr;
#pragma unroll
  for (int v = 0; v < 8; ++v) {
    const int o   = o0 + v + 8 * hi;        // D: M = vgpr + 8*hi
    const int idx = o * HW + n0 + l16;      // D: N = l16
    float r = acc[v] + bias[o];
    if (FUSED) r = g * r + resb[idx];
    Yb[idx] = r;
  }
}

// ---------------------------------------------------------------------------
// Kernel 2/4: per-row softmax stats. s[n,m] = (q[:,n].k[:,m]) / sqrt(8).
// rhd = 8, so each logit is 8 FMAs; two passes (max, then sum of exp) with
// the k-panel staged in LDS in 8x1024 tiles. One thread per query n.
// ---------------------------------------------------------------------------
__global__ __launch_bounds__(256) void attn_stats_kernel(
    const float* __restrict__ qkv, float* __restrict__ rowmax,
    float* __restrict__ rowsum)
{
  __shared__ float kl[8 * 1024];   // 32 KB
  const int tid = threadIdx.x;
  const int h   = blockIdx.y;
  const int b   = blockIdx.z;
  const int n   = blockIdx.x * 256 + tid;

  const float* base = qkv + b * (OQKV * HW);
  const float* qrow = base + (h * RHD) * HW;
  const float* krow = base + (QDIM + h * RHD) * HW;

  float q[8];
#pragma unroll
  for (int r = 0; r < 8; ++r) q[r] = qrow[r * HW + n];

  // pass 1: row max (apply 1/sqrt(8) once at the end; rs > 0)
  float mx = -3.0e38f;
  for (int m0 = 0; m0 < HW; m0 += 1024) {
    __syncthreads();
    for (int i = tid; i < 8 * 1024; i += 256)
      kl[i] = krow[(i >> 10) * HW + m0 + (i & 1023)];
    __syncthreads();
#pragma unroll 4
    for (int ml = 0; ml < 1024; ++ml) {
      float s = q[0] * kl[ml];
#pragma unroll
      for (int r = 1; r < 8; ++r) s += q[r] * kl[r * 1024 + ml];
      mx = fmaxf(mx, s);
    }
  }
  mx *= RSQRT8;

  // pass 2: denominator
  float sum = 0.0f;
  for (int m0 = 0; m0 < HW; m0 += 1024) {
    __syncthreads();
    for (int i = tid; i < 8 * 1024; i += 256)
      kl[i] = krow[(i >> 10) * HW + m0 + (i & 1023)];
    __syncthreads();
#pragma unroll 4
    for (int ml = 0; ml < 1024; ++ml) {
      float s = q[0] * kl[ml];
#pragma unroll
      for (int r = 1; r < 8; ++r) s += q[r] * kl[r * 1024 + ml];
      sum += __expf(s * RSQRT8 - mx);
    }
  }

  const int o = (b * NH + h) * HW + n;
  rowmax[o] = mx;
  rowsum[o] = sum;
}

// ---------------------------------------------------------------------------
// Kernel 3/4: O[b,h,d,n] = sum_m exp(s[n,m]-mx[n]) * v[d,m] / denom[n].
// Block = 256 threads, owns 64 queries of one (b,h). Loop m in tiles of 32:
//   step 1 (VALU): P = exp(S - mx) into LDS [32 x 64] (stride 72, so the
//                  two lane-halves of the WMMA B-read hit disjoint banks)
//   step 2 (WMMA): O += V(16xK) @ P(Kx16) per wave: 2 d-tiles x 8 K-chunks.
// Never materializes the 537 MB attention matrix in HBM.
// ---------------------------------------------------------------------------
__global__ __launch_bounds__(256) void attn_pv_kernel(
    const float* __restrict__ qkv, const float* __restrict__ rowmax,
    const float* __restrict__ rowsum, float* __restrict__ out)
{
  __shared__ float q_s[8 * 64];
  __shared__ float k_s[8 * 32];
  __shared__ float mx_s[64];
  __shared__ float p_s[32 * 72];   // [m_local][n_local], stride 72

  const int tid  = threadIdx.x;
  const int wave = tid >> 5;
  const int lane = tid & 31;
  const int hi   = lane >> 4;
  const int l16  = lane & 15;
  const int h    = blockIdx.y;
  const int b    = blockIdx.z;
  const int nb   = blockIdx.x * 64;

  const float* base = qkv + b * (OQKV * HW);
  const float* qrow = base + (h * RHD) * HW;
  const float* krow = base + (QDIM + h * RHD) * HW;
  const float* vrow = base + (2 * QDIM + h * HD) * HW;

  for (int i = tid; i < 8 * 64; i += 256)
    q_s[i] = qrow[(i >> 6) * HW + nb + (i & 63)];
  if (tid < 64) mx_s[tid] = rowmax[(b * NH + h) * HW + nb + tid];
  __syncthreads();

  const int nl = tid & 63;   // step-1 query column
  const int mg = tid >> 6;   // step-1 m group (8 m's each)
  float qv[8];
#pragma unroll
  for (int r = 0; r < 8; ++r) qv[r] = q_s[r * 64 + nl];
  const float mxv = mx_s[nl];

  const int nt = wave & 3;   // wave's 16-query tile
  const int dp = wave >> 2;  // wave's pair of 16-d tiles
  // per-lane A bases for the two d-tiles (advance by 32 floats per m-tile)
  const float* va0 = vrow + ((dp * 2 + 0) * 16 + l16) * HW + 2 * hi;
  const float* va1 = vrow + ((dp * 2 + 1) * 16 + l16) * HW + 2 * hi;
  const float* kld = krow + (tid >> 5) * HW + (tid & 31);

  v8f acc0 = {}, acc1 = {};
  for (int m0 = 0; m0 < HW; m0 += 32) {
    __syncthreads();                       // prev step-2 reads of p_s done
    k_s[tid] = kld[m0];
    __syncthreads();
    // step 1: probabilities into LDS
#pragma unroll
    for (int j = 0; j < 8; ++j) {
      const int ml = mg * 8 + j;
      float s = qv[0] * k_s[ml];
#pragma unroll
      for (int r = 1; r < 8; ++r) s += qv[r] * k_s[r * 32 + ml];
      p_s[ml * 72 + nl] = __expf(s * RSQRT8 - mxv);
    }
    __syncthreads();
    // step 2: O += V @ P on the fp32 matrix pipe
#pragma unroll
    for (int c = 0; c < 8; ++c) {
      v2f bb;
      bb.x = p_s[(4 * c + 2 * hi)     * 72 + nt * 16 + l16];
      bb.y = p_s[(4 * c + 2 * hi + 1) * 72 + nt * 16 + l16];
      v2f a0 = *(const v2f*)(va0 + m0 + 4 * c);
      v2f a1 = *(const v2f*)(va1 + m0 + 4 * c);
      acc0 = wmma_f32_16x16x4(a0, bb, acc0);
      acc1 = wmma_f32_16x16x4(a1, bb, acc1);
    }
  }

  const int   n   = nb + nt * 16 + l16;
  const float inv = 1.0f / rowsum[(b * NH + h) * HW + n];
  float* ob = out + b * (CIN * HW) + h * HD * HW + n;
#pragma unroll
  for (int v = 0; v < 8; ++v) {
    const int d0 = (dp * 2 + 0) * 16 + v + 8 * hi;
    const int d1 = (dp * 2 + 1) * 16 + v + 8 * hi;
    ob[d0 * HW] = acc0[v] * inv;
    ob[d1 * HW] = acc1[v] * inv;
  }
}

// ---------------------------------------------------------------------------
// Launch
// ---------------------------------------------------------------------------
extern "C" void kernel_launch(void* const* d_in, const int* in_sizes, int n_in,
                              void* d_out, int out_size, void* d_ws, size_t ws_size,
                              hipStream_t stream) {
  (void)n_in; (void)out_size; (void)ws_size;
  const float* x      = (const float*)d_in[0];
  const float* qkv_w  = (const float*)d_in[1];
  const float* qkv_b  = (const float*)d_in[2];
  const float* out_w  = (const float*)d_in[3];
  const float* out_b  = (const float*)d_in[4];
  const float* gamma  = (const float*)d_in[5];

  const int B = in_sizes[0] / (CIN * HW);

  float* qkv_ws  = (float*)d_ws;                  // B*320*4096
  float* attn_ws = qkv_ws  + B * (OQKV * HW);     // B*256*4096
  float* rowmax  = attn_ws + B * (CIN * HW);      // B*4*4096
  float* rowsum  = rowmax  + B * (NH * HW);       // B*4*4096

  dim3 blk(256);

  // 1) QKV projection: [320,256] @ [256,4096] + bias
  gemm_bias_kernel<false><<<dim3(HW / 64, OQKV / 32, B), blk, 0, stream>>>(
      qkv_w, x, qkv_b, qkv_ws, CIN * HW, OQKV * HW, nullptr, nullptr);

  // 2) softmax row stats (max, denom)
  attn_stats_kernel<<<dim3(HW / 256, NH, B), blk, 0, stream>>>(
      qkv_ws, rowmax, rowsum);

  // 3) fused exp(S) @ V^T with WMMA, normalized
  attn_pv_kernel<<<dim3(HW / 64, NH, B), blk, 0, stream>>>(
      qkv_ws, rowmax, rowsum, attn_ws);

  // 4) output projection + gamma*(...) + residual x
  gemm_bias_kernel<true><<<dim3(HW / 64, CIN / 32, B), blk, 0, stream>>>(
      out_w, attn_ws, out_b, (float*)d_out, CIN * HW, CIN * HW, x, gamma);
}